// DescriptorBuilder_53094385713466
// MI455X (gfx1250) — compile-verified
//
#include <hip/hip_runtime.h>

typedef __attribute__((ext_vector_type(2))) float v2f;
typedef __attribute__((ext_vector_type(8))) float v8f;

#define NATOMS 256
#define KRAD   9      // K_RADIAL+1 radial features
#define NA     4      // N_RADIAL_ANG
#define NMONO  35     // monomials deg 0..4
#define NCOLS  48     // padded to 3 WMMA N-tiles
#define OUTD   29
#define RCINV  0.2f   // 1/RC

__global__ __launch_bounds__(128) void desc_kernel(
    const float* __restrict__ R, const float* __restrict__ box,
    float* __restrict__ out)
{
  __shared__ float G_lds[NATOMS * NA];      // masked g_n(r_ij), row j
  __shared__ float U_lds[NATOMS * NCOLS];   // monomials of unit vec, row j (cols 35..47 zero)
  __shared__ float S_lds[NA * NCOLS];       // moment matrix S[n][t]
  __shared__ float qr_lds[KRAD];

  const int i   = blockIdx.x;
  const int tid = threadIdx.x;

  if (tid < KRAD) qr_lds[tid] = 0.0f;

  // ---- box and its inverse (general 3x3, adjugate) ----
  const float b0=box[0], b1=box[1], b2=box[2],
              b3=box[3], b4=box[4], b5=box[5],
              b6=box[6], b7=box[7], b8=box[8];
  const float c00 = b4*b8 - b5*b7;
  const float c01 = b5*b6 - b3*b8;
  const float c02 = b3*b7 - b4*b6;
  const float idet = 1.0f / (b0*c00 + b1*c01 + b2*c02);
  const float i00 = c00*idet, i01 = (b2*b7 - b1*b8)*idet, i02 = (b1*b5 - b2*b4)*idet;
  const float i10 = c01*idet, i11 = (b0*b8 - b2*b6)*idet, i12 = (b2*b3 - b0*b5)*idet;
  const float i20 = c02*idet, i21 = (b1*b6 - b0*b7)*idet, i22 = (b0*b4 - b1*b3)*idet;

  const float Rix = R[i*3+0], Riy = R[i*3+1], Riz = R[i*3+2];
  const float six = i00*Rix + i01*Riy + i02*Riz;
  const float siy = i10*Rix + i11*Riy + i12*Riz;
  const float siz = i20*Rix + i21*Riy + i22*Riz;

  float qr[KRAD];
  #pragma unroll
  for (int k = 0; k < KRAD; ++k) qr[k] = 0.0f;

  __syncthreads();   // qr_lds initialized

  // ---- phase 1: per-pair radial features + unit-vector monomials ----
  #pragma unroll
  for (int jj = 0; jj < 2; ++jj) {
    const int j = tid + jj*128;
    const float Rjx = R[j*3+0], Rjy = R[j*3+1], Rjz = R[j*3+2];
    const float sjx = i00*Rjx + i01*Rjy + i02*Rjz;
    const float sjy = i10*Rjx + i11*Rjy + i12*Rjz;
    const float sjz = i20*Rjx + i21*Rjy + i22*Rjz;
    float dsx = six - sjx; dsx -= rintf(dsx);   // rintf == round-half-even == jnp.round
    float dsy = siy - sjy; dsy -= rintf(dsy);
    float dsz = siz - sjz; dsz -= rintf(dsz);
    const float drx = b0*dsx + b1*dsy + b2*dsz;
    const float dry = b3*dsx + b4*dsy + b5*dsz;
    const float drz = b6*dsx + b7*dsy + b8*dsz;
    const float r2  = drx*drx + dry*dry + drz*drz;
    const float rij = sqrtf(r2 + 1e-12f);

    // Chebyshev radial basis with cosine cutoff
    const float xc = fminf(rij*RCINV, 1.0f);
    const float fc = 0.5f*(cosf(3.14159265358979323f*xc) + 1.0f);
    const float xr = rij*RCINV - 1.0f;
    const float x  = 2.0f*xr*xr - 1.0f;
    const float mask = (j == i) ? 0.0f : 1.0f;
    float phi[KRAD];
    phi[0] = fc;
    phi[1] = 0.5f*(x + 1.0f)*fc;
    float Tm2 = 1.0f, Tm1 = x;
    #pragma unroll
    for (int n = 2; n < KRAD; ++n) {
      const float Tn = 2.0f*x*Tm1 - Tm2;
      phi[n] = 0.5f*(Tn + 1.0f)*fc;
      Tm2 = Tm1; Tm1 = Tn;
    }
    #pragma unroll
    for (int k = 0; k < KRAD; ++k) qr[k] += phi[k]*mask;
    #pragma unroll
    for (int n = 0; n < NA; ++n) G_lds[j*NA + n] = phi[n]*mask;

    const float ir = 1.0f / rij;
    const float ux = drx*ir, uy = dry*ir, uz = drz*ir;
    const float xx=ux*ux, yy=uy*uy, zz=uz*uz, xy=ux*uy, xz=ux*uz, yz=uy*uz;
    float* U = &U_lds[j*NCOLS];
    U[0]=1.0f;
    U[1]=ux;     U[2]=uy;     U[3]=uz;                           // deg1 (w=1)
    U[4]=xx;     U[5]=yy;     U[6]=zz;                           // deg2 (w=1)
    U[7]=xy;     U[8]=xz;     U[9]=yz;                           // deg2 (w=2)
    U[10]=xx*ux; U[11]=yy*uy; U[12]=zz*uz;                       // deg3 (w=1)
    U[13]=xx*uy; U[14]=xx*uz; U[15]=yy*ux;                       // deg3 (w=3)
    U[16]=yy*uz; U[17]=zz*ux; U[18]=zz*uy;
    U[19]=xy*uz;                                                 // deg3 (w=6)
    U[20]=xx*xx; U[21]=yy*yy; U[22]=zz*zz;                       // deg4 (w=1)
    U[23]=xx*xy; U[24]=xx*xz; U[25]=yy*xy;                       // deg4 (w=4)
    U[26]=yy*yz; U[27]=zz*xz; U[28]=zz*yz;
    U[29]=xx*yy; U[30]=xx*zz; U[31]=yy*zz;                       // deg4 (w=6)
    U[32]=xx*yz; U[33]=yy*xz; U[34]=zz*xy;                       // deg4 (w=12)
    #pragma unroll
    for (int t = NMONO; t < NCOLS; ++t) U[t] = 0.0f;             // pad tiles
  }

  #pragma unroll
  for (int k = 0; k < KRAD; ++k) atomicAdd(&qr_lds[k], qr[k]);   // ds_add_f32

  __syncthreads();

  // ---- phase 2: S = A(16x256) x U(256x48) via WMMA f32 16x16x4 ----
  // A row m = G channel (m & 3); rows 4..15 carry duplicates -> only pollute
  // C rows 4..15, which are never read. No masking => unconditional LDS loads.
  const int wave = tid >> 5;
  const int lane = tid & 31;
  if (wave < 3) {                        // wave-uniform: EXEC all-ones inside
    const int colBase = wave * 16;       // one 16-wide N tile per wave
    const int hi2     = (lane >> 4) << 1;  // 0 (lanes 0-15) or 2 (lanes 16-31)
    // A fragment: lane L -> A[L%16, k + 2*(L>=16) + v] = G[j = k+hi2+v][L & 3]
    const float* gp = &G_lds[hi2*NA + (lane & 3)];
    // B fragment: lane L -> B[k + 2*(L>=16) + v, L%16] = U[j][colBase + L%16]
    const float* up = &U_lds[hi2*NCOLS + colBase + (lane & 15)];

    v8f acc0 = {}, acc1 = {};
    v2f a0, bv0, a1, bv1;
    // prologue: K-groups 0 and 4
    a0.x  = gp[0*NA];    a0.y  = gp[1*NA];
    bv0.x = up[0*NCOLS]; bv0.y = up[1*NCOLS];
    a1.x  = gp[4*NA];    a1.y  = gp[5*NA];
    bv1.x = up[4*NCOLS]; bv1.y = up[5*NCOLS];
    for (int k = 0; k < NATOMS - 8; k += 8) {
      v2f an0, bn0, an1, bn1;                     // prefetch groups k+8, k+12
      an0.x = gp[8*NA];     an0.y = gp[9*NA];
      bn0.x = up[8*NCOLS];  bn0.y = up[9*NCOLS];
      an1.x = gp[12*NA];    an1.y = gp[13*NA];
      bn1.x = up[12*NCOLS]; bn1.y = up[13*NCOLS];
      acc0 = __builtin_amdgcn_wmma_f32_16x16x4_f32(false, a0, false, bv0,
                                                   (short)0, acc0, false, false);
      acc1 = __builtin_amdgcn_wmma_f32_16x16x4_f32(false, a1, false, bv1,
                                                   (short)0, acc1, false, false);
      a0 = an0; bv0 = bn0; a1 = an1; bv1 = bn1;
      gp += 8*NA; up += 8*NCOLS;
    }
    acc0 = __builtin_amdgcn_wmma_f32_16x16x4_f32(false, a0, false, bv0,
                                                 (short)0, acc0, false, false);
    acc1 = __builtin_amdgcn_wmma_f32_16x16x4_f32(false, a1, false, bv1,
                                                 (short)0, acc1, false, false);
    const v8f s = acc0 + acc1;
    // C layout: VGPR v, lanes 0-15 hold row M=v -> rows 0..3 = angular channels
    if (lane < 16) {
      S_lds[0*NCOLS + colBase + lane] = s[0];
      S_lds[1*NCOLS + colBase + lane] = s[1];
      S_lds[2*NCOLS + colBase + lane] = s[2];
      S_lds[3*NCOLS + colBase + lane] = s[3];
    }
  }

  __syncthreads();

  // ---- phase 3: outputs ----
  if (tid < KRAD) out[i*OUTD + tid] = qr_lds[tid];
  if (tid >= 32 && tid < 32 + NA*5) {
    const int t20 = tid - 32;
    const int n = t20 / 5, l = t20 % 5;
    const float* S = &S_lds[n*NCOLS];
    const float T0 = S[0]*S[0];
    const float T1 = S[1]*S[1] + S[2]*S[2] + S[3]*S[3];
    const float T2 = S[4]*S[4] + S[5]*S[5] + S[6]*S[6]
                   + 2.0f*(S[7]*S[7] + S[8]*S[8] + S[9]*S[9]);
    const float T3 = S[10]*S[10] + S[11]*S[11] + S[12]*S[12]
                   + 3.0f*(S[13]*S[13] + S[14]*S[14] + S[15]*S[15]
                         + S[16]*S[16] + S[17]*S[17] + S[18]*S[18])
                   + 6.0f*S[19]*S[19];
    const float T4 = S[20]*S[20] + S[21]*S[21] + S[22]*S[22]
                   + 4.0f*(S[23]*S[23] + S[24]*S[24] + S[25]*S[25]
                         + S[26]*S[26] + S[27]*S[27] + S[28]*S[28])
                   + 6.0f*(S[29]*S[29] + S[30]*S[30] + S[31]*S[31])
                   + 12.0f*(S[32]*S[32] + S[33]*S[33] + S[34]*S[34]);
    float q;
    if      (l == 0) q = T0;
    else if (l == 1) q = T1;
    else if (l == 2) q = 1.5f*T2  - 0.5f*T0;
    else if (l == 3) q = 2.5f*T3  - 1.5f*T1;
    else             q = 4.375f*T4 - 3.75f*T2 + 0.375f*T0;
    out[i*OUTD + KRAD + t20] = q;
  }
}

extern "C" void kernel_launch(void* const* d_in, const int* in_sizes, int n_in,
                              void* d_out, int out_size, void* d_ws, size_t ws_size,
                              hipStream_t stream) {
  const float* R   = (const float*)d_in[0];   // [256,3]
  const float* box = (const float*)d_in[1];   // [3,3]
  float* out = (float*)d_out;                 // [256,29]
  (void)in_sizes; (void)n_in; (void)out_size; (void)d_ws; (void)ws_size;
  desc_kernel<<<NATOMS, 128, 0, stream>>>(R, box, out);
}